// Encoder_60713657696857
// MI455X (gfx1250) — compile-verified
//
#include <hip/hip_runtime.h>
#include <cstdint>

// Problem shape (fixed by the reference).
#define B_ 64
#define S_ 512
#define D_ 1024
#define CHUNK 4   // rows staged in LDS per iteration: 4 * 4KB = 16KB LDS

#ifndef __has_builtin
#define __has_builtin(x) 0
#endif

// ---- CDNA5 async global<->LDS data movers (gfx1250) --------------------------
// Memory-bound kernel (AI ~0.1 FLOP/byte, ~300MB traffic -> ~13us floor at
// 23.3 TB/s): the win on MI455X is wide b128 streaming through the async-LDS
// path, not matrix ops. Builtin signature (from hipcc diagnostics):
//   (v4i addrspace(1)*, v4i addrspace(3)*, imm offset, imm cpol)
#if defined(__gfx1250__) &&                                                    \
    __has_builtin(__builtin_amdgcn_global_load_async_to_lds_b128) &&           \
    __has_builtin(__builtin_amdgcn_global_store_async_from_lds_b128) &&        \
    __has_builtin(__builtin_amdgcn_s_wait_asynccnt)
#define HAVE_ASYNC 1
typedef int v4i __attribute__((vector_size(4 * sizeof(int))));
typedef __attribute__((address_space(1))) v4i GV4;
typedef __attribute__((address_space(3))) v4i LV4;
#define ASYNC_LOAD_B128(gsrc, ldst)                                            \
  __builtin_amdgcn_global_load_async_to_lds_b128((GV4*)(gsrc), (LV4*)(ldst), 0, 0)
#define ASYNC_STORE_B128(gdst, lsrc)                                           \
  __builtin_amdgcn_global_store_async_from_lds_b128((GV4*)(gdst), (LV4*)(lsrc), 0, 0)
#define WAIT_ASYNC(n) __builtin_amdgcn_s_wait_asynccnt(n)
#else
#define HAVE_ASYNC 0
#define ASYNC_LOAD_B128(gsrc, ldst)  (*(float4*)(ldst) = *(const float4*)(gsrc))
#define ASYNC_STORE_B128(gdst, lsrc) (*(float4*)(gdst) = *(const float4*)(lsrc))
#define WAIT_ASYNC(n) ((void)0)
#endif

// -----------------------------------------------------------------------------
// Kernel 1: per-row segment boundaries. word_ids is sorted & contiguous from 0,
// so word w occupies tokens [start[w], end[w]). Every w < tot_words cell read
// by kernel 2 is written here first (same stream, ordered).
// -----------------------------------------------------------------------------
__global__ void __launch_bounds__(256) seg_bounds_kernel(
    const int* __restrict__ word_ids, int* __restrict__ startw,
    int* __restrict__ endw, int* __restrict__ totw) {
  const int b = blockIdx.x;
  const int* w = word_ids + (size_t)b * S_;
  for (int t = threadIdx.x; t < S_; t += blockDim.x) {
    const int cur = w[t];
    if (t == 0 || w[t - 1] != cur) startw[b * S_ + cur] = t;
    if (t == S_ - 1) {
      endw[b * S_ + cur] = S_;
      totw[b] = cur + 1;
    } else if (w[t + 1] != cur) {
      endw[b * S_ + cur] = t + 1;
    }
  }
}

// -----------------------------------------------------------------------------
// Kernel 2: one block per output row (b, s). 256 threads x float4 = 4KB row,
// fully coalesced. Each thread only ever reads the LDS bytes it async-loaded,
// so no workgroup barrier is required around the async waits.
// -----------------------------------------------------------------------------
__global__ void __launch_bounds__(256) seg_mean_kernel(
    const float* __restrict__ h, const int* __restrict__ startw,
    const int* __restrict__ endw, const int* __restrict__ totw,
    float* __restrict__ out) {
  __shared__ __align__(16) float lds[CHUNK][D_];

  const int id = blockIdx.x;
  const int b = id >> 9;          // id / S_
  const int s = id & (S_ - 1);    // id % S_
  const int col = threadIdx.x * 4;

  const float* hb = h + (size_t)b * S_ * D_;
  float* ob = out + (size_t)b * S_ * D_;
  const int tw = totw[b];

  if (s >= tw) {
    // Pass-through row: pure DMA copy global -> LDS -> global (no VGPR data).
    const float* src = hb + (size_t)s * D_ + col;
    float* dst = ob + (size_t)s * D_ + col;
    ASYNC_LOAD_B128(src, &lds[0][col]);
    WAIT_ASYNC(0);
    ASYNC_STORE_B128(dst, &lds[0][col]);
    WAIT_ASYNC(0);
    return;
  }

  // Mean of contiguous token rows [st, en) for word w == s.
  const int st = startw[b * S_ + s];
  const int en = endw[b * S_ + s];
  const int cnt = en - st;

  float4 acc;
  acc.x = 0.f; acc.y = 0.f; acc.z = 0.f; acc.w = 0.f;

  int t = st;
  while (t < en) {
    const int n = (en - t < CHUNK) ? (en - t) : CHUNK;
    // Issue all async row loads for this chunk, then drain and accumulate.
    for (int i = 0; i < n; ++i)
      ASYNC_LOAD_B128(hb + (size_t)(t + i) * D_ + col, &lds[i][col]);
    WAIT_ASYNC(0);
    for (int i = 0; i < n; ++i) {
      const float4 v = *(const float4*)&lds[i][col];
      acc.x += v.x; acc.y += v.y; acc.z += v.z; acc.w += v.w;
    }
    t += n;
  }

  const float inv = 1.0f / (float)cnt;
  float4 r;
  r.x = acc.x * inv; r.y = acc.y * inv; r.z = acc.z * inv; r.w = acc.w * inv;
  *(float4*)(ob + (size_t)s * D_ + col) = r;
}

// -----------------------------------------------------------------------------
extern "C" void kernel_launch(void* const* d_in, const int* in_sizes, int n_in,
                              void* d_out, int out_size, void* d_ws, size_t ws_size,
                              hipStream_t stream) {
  (void)in_sizes; (void)n_in; (void)out_size; (void)ws_size;

  const float* h = (const float*)d_in[0];      // [B, S, D] fp32
  const int* word_ids = (const int*)d_in[1];   // [B, S] int32
  float* out = (float*)d_out;                  // [B, S, D] fp32

  int* startw = (int*)d_ws;                    // B*S ints
  int* endw   = startw + B_ * S_;              // B*S ints
  int* totw   = endw + B_ * S_;                // B ints

  seg_bounds_kernel<<<B_, 256, 0, stream>>>(word_ids, startw, endw, totw);
  seg_mean_kernel<<<B_ * S_, 256, 0, stream>>>(h, startw, endw, totw, out);
}